// TransformFeatureQuaternion_41334765256759
// MI455X (gfx1250) — compile-verified
//
#include <hip/hip_runtime.h>
#include <hip/hip_bf16.h>

typedef __attribute__((ext_vector_type(2))) float v2f;
typedef __attribute__((ext_vector_type(4))) float v4f;
typedef __attribute__((ext_vector_type(8))) float v8f;

#define DIM      1024
#define QPTS     4096
#define TROT     64
#define MULC     64
#define QT       16      // q rows per block tile
#define NTHREADS 256
#define NWAVES   8
#define NTILES   ((QT * MULC) / 16)        // 64 WMMA tiles per l per block
#define NITER    (NTILES / NWAVES)         // 8 tiles per wave

// Wigner-D block per rotation: 1 + 9 + 25 + 49 = 84 floats
#define DBLK  84
#define DOFF0 0
#define DOFF1 1
#define DOFF2 10
#define DOFF3 35
#define BASE0 0
#define BASE1 64
#define BASE2 256
#define BASE3 576

// ---------------------------------------------------------------------------
// Kernel 1: build Wigner D matrices (one thread per rotation)
// ---------------------------------------------------------------------------
__device__ inline void zrot(float ang, int l, float* X) {
  int d = 2 * l + 1;
  for (int a = 0; a < d * d; ++a) X[a] = 0.f;
  for (int kk = 0; kk < d; ++kk)                 // sin on anti-diagonal first
    X[kk * d + (d - 1 - kk)] = sinf((float)(l - kk) * ang);
  for (int kk = 0; kk < d; ++kk)                 // cos overwrites center (k=l)
    X[kk * d + kk] = cosf((float)(l - kk) * ang);
}

__device__ inline void matmul(const float* A, const float* B, float* C, int d) {
  for (int i = 0; i < d; ++i)
    for (int j = 0; j < d; ++j) {
      float s = 0.f;
      for (int k = 0; k < d; ++k) s += A[i * d + k] * B[k * d + j];
      C[i * d + j] = s;
    }
}

__global__ void build_wigner_kernel(const float* __restrict__ q,
                                    const float* __restrict__ J0,
                                    const float* __restrict__ J1,
                                    const float* __restrict__ J2,
                                    const float* __restrict__ J3,
                                    float* __restrict__ Dws) {
  int t = threadIdx.x;
  if (t >= TROT) return;
  float w = q[t * 4 + 0], x = q[t * 4 + 1], y = q[t * 4 + 2], z = q[t * 4 + 3];
  float n = sqrtf(w * w + x * x + y * y + z * z);
  w /= n; x /= n; y /= n; z /= n;
  if (w < 0.f) { w = -w; x = -x; y = -y; z = -z; }
  float two_s = 2.f / (w * w + x * x + y * y + z * z);
  float M01 = two_s * (x * y - z * w);
  float M21 = two_s * (y * z + x * w);
  float M11 = 1.f - two_s * (x * x + z * z);
  float M10 = two_s * (x * y + z * w);
  float M12 = two_s * (y * z - x * w);
  float alpha = atan2f(M01, M21);
  float beta  = acosf(fminf(1.f, fmaxf(-1.f, M11)));
  float gamma = atan2f(M10, -M12);

  const float* Js[4]   = {J0, J1, J2, J3};
  const int   doffs[4] = {DOFF0, DOFF1, DOFF2, DOFF3};
  float Xr[49], T1[49], T2[49];
  float* Dt = Dws + t * DBLK;
  for (int l = 0; l < 4; ++l) {
    int d = 2 * l + 1;
    const float* J = Js[l];
    zrot(alpha, l, Xr); matmul(Xr, J, T1, d);   // Xa @ J
    zrot(beta,  l, Xr); matmul(T1, Xr, T2, d);  // .. @ Xb
    matmul(T2, J, T1, d);                       // .. @ J
    zrot(gamma, l, Xr); matmul(T1, Xr, T2, d);  // .. @ Xc
    for (int a = 0; a < d * d; ++a) Dt[doffs[l] + a] = T2[a];
  }
}

// ---------------------------------------------------------------------------
// Kernel 2: apply block-diagonal D with f32 WMMA (16x16x4), LDS-staged output
// ---------------------------------------------------------------------------
template <int D_, int BASE_, int DOFF_>
__device__ inline void wmma_block(const float* __restrict__ feature,
                                  float* __restrict__ sOut,
                                  const float* __restrict__ sD,
                                  int q0, int waveId, int lane) {
  constexpr int KSTEPS = (D_ + 3) / 4;
  const int hi = lane >> 4;   // 0: K slots {0,1}/{4,5}; 1: K slots {2,3}/{6,7}
  const int li = lane & 15;

  // Loop-invariant A operand (D matrix rows i=li, zero-padded to 16x4/K-step).
  // Zeroed A columns for j >= D_ also nullify any garbage in B's padded rows,
  // so B needs no masking at all.
  v2f a[KSTEPS];
#pragma unroll
  for (int s = 0; s < KSTEPS; ++s) {
    const int j0  = 4 * s + 2 * hi;
    const int j1  = j0 + 1;
    const int j0c = (j0 < D_) ? j0 : 0;   // clamped: stays inside sD[160]
    const int j1c = (j1 < D_) ? j1 : 0;
    float a0 = sD[DOFF_ + li * D_ + j0c];
    float a1 = sD[DOFF_ + li * D_ + j1c];
    a[s].x = (li < D_ && j0 < D_) ? a0 : 0.f;
    a[s].y = (li < D_ && j1 < D_) ? a1 : 0.f;
  }

  // Per-lane 32-bit feature offsets (element units). Wave w starts at tile w;
  // stepping tiles by NWAVES advances the pair index by 128 = 2 full q rows,
  // leaving m unchanged -> every offset just grows by 2*DIM per iteration.
  const int p  = waveId * 16 + li;
  const int ql = p >> 6;
  const int m  = p & 63;
  const int rowoff = (q0 + ql) * DIM + BASE_ + m * D_;
  int boff[KSTEPS][2];
#pragma unroll
  for (int s = 0; s < KSTEPS; ++s) {
    const int j0 = 4 * s + 2 * hi;
    const int j1 = j0 + 1;
    boff[s][0] = rowoff + ((j0 < D_) ? j0 : 0);  // clamped in-bounds; garbage
    boff[s][1] = rowoff + ((j1 < D_) ? j1 : 0);  // killed by zeroed A column
  }
  int sAddr = ql * DIM + BASE_ + m * D_;         // LDS store offset (lanes<16)
  const bool isLo = (lane < 16);

#pragma unroll
  for (int it = 0; it < NITER; ++it) {
    v8f acc = {0.f, 0.f, 0.f, 0.f, 0.f, 0.f, 0.f, 0.f};
#pragma unroll
    for (int s = 0; s < KSTEPS; ++s) {
      v2f b;
      b.x = feature[boff[s][0]];
      b.y = feature[boff[s][1]];
      acc = __builtin_amdgcn_wmma_f32_16x16x4_f32(false, a[s], false, b,
                                                  (short)0, acc, false, false);
    }
    // C/D tile: VGPR r, lanes 0-15 -> row i=r, col=pair (lanes 16-31 carry
    // rows 8..15, which are zero since A rows >= D_ were zero-padded).
    if (isLo) {
#pragma unroll
      for (int r = 0; r < D_; ++r) sOut[sAddr + r] = acc[r];
    }
#pragma unroll
    for (int s = 0; s < KSTEPS; ++s) {
      boff[s][0] += 2 * DIM;
      boff[s][1] += 2 * DIM;
    }
    sAddr += 2 * DIM;
  }
}

__global__ void __launch_bounds__(NTHREADS)
transform_kernel(const float* __restrict__ feature,
                 const float* __restrict__ Dws,
                 float* __restrict__ out) {
  __shared__ float sD[160];          // 84 used; oversized so clamped A-reads
                                     // from rows i>=d stay inside the array
  __shared__ float sOut[QT * DIM];   // 64 KB output tile
  const int tid = threadIdx.x;
  const int q0  = blockIdx.x * QT;
  const int t   = blockIdx.y;

  if (tid < 160) sD[tid] = (tid < DBLK) ? Dws[t * DBLK + tid] : 0.f;
  __syncthreads();

  // l = 0: 1x1 block -> scaled copy of first 64 channels
  {
    const float d0 = sD[DOFF0];
    for (int idx = tid; idx < QT * MULC; idx += NTHREADS) {
      int ql = idx >> 6, c = idx & 63;
      sOut[ql * DIM + c] = d0 * feature[(size_t)(q0 + ql) * DIM + BASE0 + c];
    }
  }

  const int waveId = tid >> 5;
  const int lane   = tid & 31;
  wmma_block<3, BASE1, DOFF1>(feature, sOut, sD, q0, waveId, lane);
  wmma_block<5, BASE2, DOFF2>(feature, sOut, sD, q0, waveId, lane);
  wmma_block<7, BASE3, DOFF3>(feature, sOut, sD, q0, waveId, lane);
  __syncthreads();

  // Flush the contiguous 64 KB slab: coalesced 128-bit nontemporal stores
  // (don't let the 1 GiB output stream evict the L2-resident feature tensor)
  const size_t obase = ((size_t)t * QPTS + q0) * DIM;
  const v4f* src = (const v4f*)sOut;
  v4f* dst = (v4f*)(out + obase);
#pragma unroll
  for (int idx = tid; idx < (QT * DIM) / 4; idx += NTHREADS) {
    __builtin_nontemporal_store(src[idx], &dst[idx]);
  }
}

// ---------------------------------------------------------------------------
extern "C" void kernel_launch(void* const* d_in, const int* in_sizes, int n_in,
                              void* d_out, int out_size, void* d_ws, size_t ws_size,
                              hipStream_t stream) {
  (void)in_sizes; (void)n_in; (void)out_size; (void)ws_size;
  const float* feature = (const float*)d_in[0];
  const float* q       = (const float*)d_in[1];
  const float* J0      = (const float*)d_in[2];
  const float* J1      = (const float*)d_in[3];
  const float* J2      = (const float*)d_in[4];
  const float* J3      = (const float*)d_in[5];
  float* out = (float*)d_out;
  float* Dws = (float*)d_ws;   // 64 * 84 floats

  build_wigner_kernel<<<1, 64, 0, stream>>>(q, J0, J1, J2, J3, Dws);

  dim3 grid(QPTS / QT, TROT);  // (256, 64)
  transform_kernel<<<grid, NTHREADS, 0, stream>>>(feature, Dws, out);
}